// FullSeqModel_84078279787064
// MI455X (gfx1250) — compile-verified
//
#include <hip/hip_runtime.h>
#include <math.h>

// ---------------------------------------------------------------------------
// Tiny-transformer autoregressive decode for MI455X (gfx1250, wave32).
// Persistent-kernel design: 16 blocks x 256 threads (=128 wave32s), one-shot
// global barriers in scratch (re-zeroed by k_init each call -> graph-safe).
// FFN GEMMs use V_WMMA_F32_16X16X4_F32 (K=4 pads D=3, full fp32 numerics).
// FFN weights are staged per-block into LDS with GLOBAL_LOAD_ASYNC_TO_LDS_B128
// (ASYNCcnt) -> 8x less L2 traffic and branch-free WMMA operand fetch.
// ---------------------------------------------------------------------------

#define NBLK   16
#define NTHR   256
#define NSLOT  6144        // >= 4715 barriers actually used per call
#define TKN    2048        // T*B tokens
#define FFN    2048
#define NLAY   12
#define TT     64
#define BBATCH 32
#define LNEPS  1e-5f

typedef float v2f __attribute__((ext_vector_type(2)));
typedef float v8f __attribute__((ext_vector_type(8)));

#if defined(__HIP_DEVICE_COMPILE__)
#define WAIT_DS()    asm volatile("s_wait_dscnt 0" ::: "memory")
#define WAIT_ASYNC() asm volatile("s_wait_asynccnt 0" ::: "memory")
#define ASYNC_LDS_B128(ldsoff, gaddr)                                          \
  asm volatile("global_load_async_to_lds_b128 %0, %1, off"                     \
               :: "v"(ldsoff), "v"(gaddr) : "memory")
#else
#define WAIT_DS()
#define WAIT_ASYNC()
#define ASYNC_LDS_B128(ldsoff, gaddr)
#endif

struct Params {
  const float *src, *angle;
  const float *e_aw, *e_ab, *e_aow, *e_aob;
  const float *e_g1, *e_b1, *e_g2, *e_b2;
  const float *e_f1w, *e_f1b, *e_f2w, *e_f2b;
  const float *e_ng, *e_nb;
  const float *d_sw, *d_sb, *d_sow, *d_sob;
  const float *d_cw, *d_cb, *d_cow, *d_cob;
  const float *d_g1, *d_b1, *d_g2, *d_b2, *d_g3, *d_b3;
  const float *d_f1w, *d_f1b, *d_f2w, *d_f2b;
  const float *d_ng, *d_nb;
  float *out;
  unsigned int *slots;
  float *arena;
};

__global__ void k_init(unsigned int* slots) {
  for (int i = threadIdx.x; i < NSLOT; i += blockDim.x) slots[i] = 0u;
}

// One-shot slot barrier: each slot used exactly once per call, zeroed by k_init.
__device__ __forceinline__ void gbarrier(unsigned int* slots, int& bi) {
  __syncthreads();
  __threadfence();
  if (threadIdx.x == 0) {
    unsigned int* s = slots + bi;
    __hip_atomic_fetch_add(s, 1u, __ATOMIC_ACQ_REL, __HIP_MEMORY_SCOPE_AGENT);
    while (__hip_atomic_load(s, __ATOMIC_ACQUIRE, __HIP_MEMORY_SCOPE_AGENT) <
           (unsigned)NBLK)
      __builtin_amdgcn_s_sleep(1);
  }
  __builtin_amdgcn_s_cluster_barrier();  // NOP when not dispatched as a cluster
  __syncthreads();
  ++bi;
}

// D = A(16x4) * B(4x16) + C, fp32.  A: lane<16 {K0,K1}, lane>=16 {K2,K3}.
// B assumed symmetric: v0 = row K0 (lanes 0-15) / K2 (lanes 16-31); v1 = K1/K3.
__device__ __forceinline__ v8f wmma4(v2f a, v2f b, v8f c) {
  return __builtin_amdgcn_wmma_f32_16x16x4_f32(false, a, false, b, (short)0, c,
                                               false, false);
}

__device__ __forceinline__ void ln_store(float v0, float v1, float v2,
                                         const float* g, const float* b,
                                         float* o) {
  float m  = (v0 + v1 + v2) * (1.f / 3.f);
  float d0 = v0 - m, d1 = v1 - m, d2 = v2 - m;
  float inv = 1.f / sqrtf((d0 * d0 + d1 * d1 + d2 * d2) * (1.f / 3.f) + LNEPS);
  o[0] = d0 * inv * g[0] + b[0];
  o[1] = d1 * inv * g[1] + b[1];
  o[2] = d2 * inv * g[2] + b[2];
}

__device__ __forceinline__ void qkv_phase(int gtid, const float* __restrict__ h,
                                          const float* __restrict__ w,
                                          const float* __restrict__ b,
                                          float* q, float* k, float* v) {
  if (gtid < TKN) {
    const float* hr = h + gtid * 3;
    float h0 = hr[0], h1 = hr[1], h2 = hr[2];
#pragma unroll
    for (int j = 0; j < 3; ++j) {
      q[gtid*3+j] = b[j]   + w[j*3+0]*h0     + w[j*3+1]*h1     + w[j*3+2]*h2;
      k[gtid*3+j] = b[3+j] + w[(3+j)*3+0]*h0 + w[(3+j)*3+1]*h1 + w[(3+j)*3+2]*h2;
      v[gtid*3+j] = b[6+j] + w[(6+j)*3+0]*h0 + w[(6+j)*3+1]*h1 + w[(6+j)*3+2]*h2;
    }
  }
}

__device__ __forceinline__ void attn_core(float qs, const float* __restrict__ kk,
                                          const float* __restrict__ vv,
                                          int bcol, int hh, float* dst) {
  float mx = -3.4e38f;                          // head dim 1 -> scale = 1
  for (int s = 0; s < TT; ++s)
    mx = fmaxf(mx, qs * kk[(s * BBATCH + bcol) * 3 + hh]);
  float se = 0.f, acc = 0.f;
  for (int s = 0; s < TT; ++s) {
    float e = expf(qs * kk[(s * BBATCH + bcol) * 3 + hh] - mx);
    se  += e;
    acc += e * vv[(s * BBATCH + bcol) * 3 + hh];
  }
  *dst = acc / se;
}

__device__ __forceinline__ void self_attn_phase(int gtid, const float* q,
                                                const float* k, const float* v,
                                                float* ao) {
  if (gtid < TKN * 3) {
    int hh = gtid % 3, bcol = (gtid / 3) & 31, qp = gtid / 96;
    float qs = q[(qp * BBATCH + bcol) * 3 + hh];
    attn_core(qs, k, v, bcol, hh, &ao[(qp * BBATCH + bcol) * 3 + hh]);
  }
}

__device__ __forceinline__ void cross_attn_phase(int gtid, const float* h,
                                                 const float* cw, const float* cb,
                                                 const float* km, const float* vm,
                                                 float* ao) {
  if (gtid < TKN * 3) {
    int hh = gtid % 3, bcol = (gtid / 3) & 31, qp = gtid / 96;
    const float* hr = h + (qp * BBATCH + bcol) * 3;
    float qs = cb[hh] + cw[hh*3+0]*hr[0] + cw[hh*3+1]*hr[1] + cw[hh*3+2]*hr[2];
    attn_core(qs, km, vm, bcol, hh, &ao[(qp * BBATCH + bcol) * 3 + hh]);
  }
}

__device__ __forceinline__ void proj_ln_phase(int gtid, float* h, const float* ao,
                                              const float* ow, const float* obv,
                                              const float* g, const float* b) {
  if (gtid < TKN) {
    const float* o = ao + gtid * 3;
    float* hr = h + gtid * 3;
    float p0 = obv[0] + ow[0]*o[0] + ow[1]*o[1] + ow[2]*o[2];
    float p1 = obv[1] + ow[3]*o[0] + ow[4]*o[1] + ow[5]*o[2];
    float p2 = obv[2] + ow[6]*o[0] + ow[7]*o[1] + ow[8]*o[2];
    ln_store(hr[0] + p0, hr[1] + p1, hr[2] + p2, g, b, hr);
  }
}

// FFN (x@W1^T+b1 -> relu -> @W2^T+b2) + residual + LN, one wave per 16 tokens.
// WMMA f32 16x16x4: K padded 3->4 (GEMM1) and FF chunked by 4 (GEMM2).
// W1/W2 staged into LDS per block via async DMA (ASYNCcnt path).
__device__ void ffn_ln_phase(int gtid, float* __restrict__ h,
                             const float* __restrict__ w1, const float* __restrict__ b1,
                             const float* __restrict__ w2, const float* __restrict__ b2,
                             const float* g, const float* bb,
                             float* w1s, float* w2s, float* Hs, float* Ys) {
  // ---- stage W1 [FFN][3] and W2 [3][FFN] (24KB each) into LDS, async ----
  {
    const unsigned w1o = (unsigned)(unsigned long long)(const void*)w1s;
    const unsigned w2o = (unsigned)(unsigned long long)(const void*)w2s;
    const unsigned long long g1 = (unsigned long long)(const void*)w1;
    const unsigned long long g2 = (unsigned long long)(const void*)w2;
    const unsigned tid = threadIdx.x;
#pragma unroll
    for (int i = 0; i < 6; ++i) {              // 1536 x 16B chunks per matrix
      unsigned byteoff = (tid + (unsigned)i * 256u) * 16u;
      ASYNC_LDS_B128(w1o + byteoff, g1 + (unsigned long long)byteoff);
      ASYNC_LDS_B128(w2o + byteoff, g2 + (unsigned long long)byteoff);
    }
    WAIT_ASYNC();
    __syncthreads();
  }

  const int lane = threadIdx.x & 31;
  const int gw   = gtid >> 5;       // 0..127 == m-tile (grid has exactly 128 waves)
  const int m0   = gw << 4;
  const int rw   = lane & 15;
  const bool hi  = lane >= 16;

  __builtin_prefetch(b1, 0, 3);

  v2f A;
  {
    const float* hr = h + (m0 + rw) * 3;
    A.x = hi ? hr[2] : hr[0];
    A.y = hi ? 0.0f  : hr[1];       // K=3 pad column
  }

  v8f Cy = {0.f, 0.f, 0.f, 0.f, 0.f, 0.f, 0.f, 0.f};
  for (int n = 0; n < 128; ++n) {
    const int f0 = n << 4;
    const int fc = f0 + rw;
    // B1[k][col] = W1[f0+col][k], k<3 ; row K=3 is zero.  Branch-free LDS reads.
    const float* w1r = w1s + fc * 3;
    float b1x = w1r[hi ? 2 : 0];
    float b1y = w1r[1];
    v2f B1;
    B1.x = b1x;
    B1.y = hi ? 0.0f : b1y;
    v8f C1 = {0.f, 0.f, 0.f, 0.f, 0.f, 0.f, 0.f, 0.f};
    C1 = wmma4(A, B1, C1);

    // bias + relu in C layout (this lane owns ff column fc), stage to LDS
    const float bias  = b1[fc];
    const int   rbase = hi ? 8 : 0;
#pragma unroll
    for (int r = 0; r < 8; ++r) {
      float xv = C1[r] + bias;
      Hs[(rbase + r) * 16 + rw] = xv > 0.f ? xv : 0.f;
    }
    WAIT_DS();                       // LDS is in-order per wave; block reordering
    // consume the 16-ff tile as 4 K=4 WMMAs of GEMM2
    const int rcl = rw < 3 ? rw : 0; // clamped row; masked below
#pragma unroll
    for (int g4 = 0; g4 < 4; ++g4) {
      const float* hr2 = Hs + rw * 16 + (g4 << 2) + (hi ? 2 : 0);
      v2f A2 = *(const v2f*)hr2;     // 8B-aligned ds_load_b64
      const int fk = f0 + (g4 << 2) + (hi ? 2 : 0);
      float t0 = w2s[rcl * FFN + fk + 0];
      float t1 = w2s[rcl * FFN + fk + 1];
      v2f B2;
      B2.x = (rw < 3) ? t0 : 0.f;
      B2.y = (rw < 3) ? t1 : 0.f;
      Cy = wmma4(A2, B2, Cy);
    }
    WAIT_DS();
  }

  // Extract Y (cols 0..2), + b2, stage per-token rows to LDS
  if (rw < 3) {
    const float badd  = b2[rw];
    const int   rbase = hi ? 8 : 0;
#pragma unroll
    for (int r = 0; r < 8; ++r) Ys[(rbase + r) * 4 + rw] = Cy[r] + badd;
  }
  WAIT_DS();
  if (lane < 16) {                   // one token per lane: residual + LN
    const int i = m0 + lane;
    float v0 = h[i * 3 + 0] + Ys[lane * 4 + 0];
    float v1 = h[i * 3 + 1] + Ys[lane * 4 + 1];
    float v2 = h[i * 3 + 2] + Ys[lane * 4 + 2];
    ln_store(v0, v1, v2, g, bb, h + i * 3);
  }
}

__global__ void __launch_bounds__(NTHR) k_main(Params P) {
  __shared__ __attribute__((aligned(16))) float w1s[FFN * 3];  // 24KB
  __shared__ __attribute__((aligned(16))) float w2s[3 * FFN];  // 24KB
  __shared__ __attribute__((aligned(16))) float ldsH[8 * 256]; // 8KB
  __shared__ __attribute__((aligned(16))) float ldsY[8 * 64];  // 2KB
  const int gtid = (int)(blockIdx.x * NTHR + threadIdx.x);
  int bi = 0;

  float* x    = P.arena;
  float* mem  = x    + TKN * 3;
  float* h    = mem  + TKN * 3;
  float* q    = h    + TKN * 3;
  float* k    = q    + TKN * 3;
  float* v    = k    + TKN * 3;
  float* ao   = v    + TKN * 3;
  float* outb = ao   + TKN * 3;
  float* km   = outb + TKN * 3;              // [NLAY][TKN*3]
  float* vm   = km   + NLAY * TKN * 3;
  float* Hs   = ldsH + (threadIdx.x >> 5) * 256;
  float* Ys   = ldsY + (threadIdx.x >> 5) * 64;

  // x[t,b,:] = {src[b,0,t], src[b,1,t], angle[b]}
  if (gtid < TKN) {
    int b = gtid & 31, t = gtid >> 5;
    x[gtid * 3 + 0] = P.src[b * 2 * TT + t];
    x[gtid * 3 + 1] = P.src[b * 2 * TT + TT + t];
    x[gtid * 3 + 2] = P.angle[b];
  }
  gbarrier(P.slots, bi);
  if (gtid < TKN * 3) h[gtid] = x[gtid];
  gbarrier(P.slots, bi);

  // -------- encoder (once) --------
  for (int l = 0; l < NLAY; ++l) {
    qkv_phase(gtid, h, P.e_aw + l * 27, P.e_ab + l * 9, q, k, v);
    gbarrier(P.slots, bi);
    self_attn_phase(gtid, q, k, v, ao);
    gbarrier(P.slots, bi);
    proj_ln_phase(gtid, h, ao, P.e_aow + l * 9, P.e_aob + l * 3,
                  P.e_g1 + l * 3, P.e_b1 + l * 3);
    gbarrier(P.slots, bi);
    ffn_ln_phase(gtid, h, P.e_f1w + l * FFN * 3, P.e_f1b + l * FFN,
                 P.e_f2w + l * 3 * FFN, P.e_f2b + l * 3,
                 P.e_g2 + l * 3, P.e_b2 + l * 3, w1s, w2s, Hs, Ys);
    gbarrier(P.slots, bi);
  }
  if (gtid < TKN) {
    const float* hr = h + gtid * 3;
    ln_store(hr[0], hr[1], hr[2], P.e_ng, P.e_nb, mem + gtid * 3);
  }
  gbarrier(P.slots, bi);

  // cross-attn K/V depend only on mem -> precompute per layer once
  for (int idx = gtid; idx < NLAY * TKN; idx += NBLK * NTHR) {
    int l = idx >> 11, tok = idx & (TKN - 1);
    const float* w  = P.d_cw + l * 27;
    const float* bv = P.d_cb + l * 9;
    const float* mr = mem + tok * 3;
#pragma unroll
    for (int j = 0; j < 3; ++j) {
      km[l*TKN*3 + tok*3 + j] = bv[3+j] + w[(3+j)*3+0]*mr[0] + w[(3+j)*3+1]*mr[1] + w[(3+j)*3+2]*mr[2];
      vm[l*TKN*3 + tok*3 + j] = bv[6+j] + w[(6+j)*3+0]*mr[0] + w[(6+j)*3+1]*mr[1] + w[(6+j)*3+2]*mr[2];
    }
  }
  gbarrier(P.slots, bi);

  // out = zeros.at[0].set(x[0])
  if (gtid < TKN * 3) outb[gtid] = (gtid < BBATCH * 3) ? x[gtid] : 0.f;
  gbarrier(P.slots, bi);

  // -------- autoregressive decode: 63 full decoder passes --------
  for (int t = 1; t < TT; ++t) {
    if (gtid < TKN * 3) h[gtid] = outb[gtid];
    gbarrier(P.slots, bi);
    for (int l = 0; l < NLAY; ++l) {
      qkv_phase(gtid, h, P.d_sw + l * 27, P.d_sb + l * 9, q, k, v);
      gbarrier(P.slots, bi);
      self_attn_phase(gtid, q, k, v, ao);
      gbarrier(P.slots, bi);
      proj_ln_phase(gtid, h, ao, P.d_sow + l * 9, P.d_sob + l * 3,
                    P.d_g1 + l * 3, P.d_b1 + l * 3);
      gbarrier(P.slots, bi);
      cross_attn_phase(gtid, h, P.d_cw + l * 27, P.d_cb + l * 9,
                       km + l * TKN * 3, vm + l * TKN * 3, ao);
      gbarrier(P.slots, bi);
      proj_ln_phase(gtid, h, ao, P.d_cow + l * 9, P.d_cob + l * 3,
                    P.d_g2 + l * 3, P.d_b2 + l * 3);
      gbarrier(P.slots, bi);
      ffn_ln_phase(gtid, h, P.d_f1w + l * FFN * 3, P.d_f1b + l * FFN,
                   P.d_f2w + l * 3 * FFN, P.d_f2b + l * 3,
                   P.d_g3 + l * 3, P.d_b3 + l * 3, w1s, w2s, Hs, Ys);
      gbarrier(P.slots, bi);
    }
    if (gtid < TKN && (gtid >> 5) == t) {      // keep only position t
      const float* hr = h + gtid * 3;
      ln_store(hr[0], hr[1], hr[2], P.d_ng, P.d_nb, outb + gtid * 3);
    }
    gbarrier(P.slots, bi);
  }

  // transpose to [B,3,T]
  if (gtid < TKN * 3) {
    int b = gtid / 192, r = gtid % 192, c = r / 64, t = r % 64;
    P.out[gtid] = outb[(t * BBATCH + b) * 3 + c];
  }
}

extern "C" void kernel_launch(void* const* d_in, const int* in_sizes, int n_in,
                              void* d_out, int out_size, void* d_ws, size_t ws_size,
                              hipStream_t stream) {
  (void)in_sizes; (void)n_in; (void)out_size; (void)ws_size;
  Params P;
  const float** fp = (const float**)d_in;
  P.src  = fp[0];  P.angle = fp[1];
  P.e_aw = fp[2];  P.e_ab = fp[3];  P.e_aow = fp[4];  P.e_aob = fp[5];
  P.e_g1 = fp[6];  P.e_b1 = fp[7];  P.e_g2  = fp[8];  P.e_b2  = fp[9];
  P.e_f1w = fp[10]; P.e_f1b = fp[11]; P.e_f2w = fp[12]; P.e_f2b = fp[13];
  P.e_ng = fp[14]; P.e_nb = fp[15];
  P.d_sw = fp[16]; P.d_sb = fp[17]; P.d_sow = fp[18]; P.d_sob = fp[19];
  P.d_cw = fp[20]; P.d_cb = fp[21]; P.d_cow = fp[22]; P.d_cob = fp[23];
  P.d_g1 = fp[24]; P.d_b1 = fp[25]; P.d_g2 = fp[26]; P.d_b2 = fp[27];
  P.d_g3 = fp[28]; P.d_b3 = fp[29];
  P.d_f1w = fp[30]; P.d_f1b = fp[31]; P.d_f2w = fp[32]; P.d_f2b = fp[33];
  P.d_ng = fp[34]; P.d_nb = fp[35];
  P.out   = (float*)d_out;
  P.slots = (unsigned int*)d_ws;
  P.arena = (float*)((char*)d_ws + NSLOT * sizeof(unsigned int));

  hipLaunchKernelGGL(k_init, dim3(1), dim3(NTHR), 0, stream, P.slots);
  hipLaunchKernelGGL(k_main, dim3(NBLK), dim3(NTHR), 0, stream, P);
}